// ARX_LeafRiver_Qsim_80693845557569
// MI455X (gfx1250) — compile-verified
//
#include <hip/hip_runtime.h>

// out[b] = dot(x[b, 0, 0:8], weight[0:8]) + bias   (weight_y term is exactly zero)
// Memory-bound strided gather; dot product mapped onto V_WMMA_F32_16X16X4_F32.

typedef __attribute__((ext_vector_type(2))) float v2f;
typedef __attribute__((ext_vector_type(8))) float v8f;

static constexpr int kBatch     = 65536;
static constexpr int kSeq       = 365;
static constexpr int kIn        = 8;
static constexpr int kRowStride = kSeq * kIn;   // floats between consecutive batch rows of x

static constexpr int kWavesPerBlock = 8;                     // 256 threads = 8 wave32
static constexpr int kRowsPerBlock  = kWavesPerBlock * 16;   // one 16-row WMMA tile per wave

__global__ __launch_bounds__(kWavesPerBlock * 32)
void ARX_LeafRiver_Qsim_kernel(const float* __restrict__ x,
                               const float* __restrict__ w,
                               const float* __restrict__ bias,
                               float* __restrict__ out)
{
    const int lane    = threadIdx.x & 31;
    const int wave    = threadIdx.x >> 5;
    const int rowBase = (blockIdx.x * kWavesPerBlock + wave) * 16;

    const int m     = lane & 15;        // M index within the 16-row tile
    const int khalf = (lane >> 4) * 2;  // lanes 0-15 -> K base 0, lanes 16-31 -> K base 2

    // A-matrix (16x4 f32): VGPR0 = K=khalf, VGPR1 = K=khalf+1, per ISA 7.12.2.
    // One b64 load per lane per WMMA = exactly the 32 bytes/row the problem needs.
    const float* xr = x + (size_t)(rowBase + m) * kRowStride + khalf;
    v2f a0 = *(const v2f*)(xr);        // covers K = 0..3 across the two lane halves
    v2f a1 = *(const v2f*)(xr + 4);    // covers K = 4..7

    // B-matrix (4x16 f32): row K broadcast across N (all lanes in a half hold w[K]).
    // Only column N=0 of D is consumed, so replication across N is harmless.
    v2f b0, b1;
    b0.x = w[khalf + 0]; b0.y = w[khalf + 1];
    b1.x = w[khalf + 4]; b1.y = w[khalf + 5];

    // Fold the bias add into the accumulator init.
    const float bv = bias[0];
    v8f c = {bv, bv, bv, bv, bv, bv, bv, bv};

    // Two chained v_wmma_f32_16x16x4_f32 (K = 0..3, then 4..7).
    // 8-arg form: (neg_a, A, neg_b, B, c_mod, C, reuse_a, reuse_b)
    c = __builtin_amdgcn_wmma_f32_16x16x4_f32(false, a0, false, b0, (short)0, c, false, false);
    c = __builtin_amdgcn_wmma_f32_16x16x4_f32(false, a1, false, b1, (short)0, c, false, false);

    // D[M, N=0]: lane 0 holds M=0..7 in c[0..7]; lane 16 holds M=8..15 (ISA 7.12.2).
    // Those two lanes write 8 consecutive fp32 each as two b128 stores.
    if (m == 0) {
        float* o = out + rowBase + (lane >> 4) * 8;
        *(float4*)(o + 0) = make_float4(c[0], c[1], c[2], c[3]);
        *(float4*)(o + 4) = make_float4(c[4], c[5], c[6], c[7]);
    }
}

extern "C" void kernel_launch(void* const* d_in, const int* in_sizes, int n_in,
                              void* d_out, int out_size, void* d_ws, size_t ws_size,
                              hipStream_t stream) {
    const float* x    = (const float*)d_in[0];  // (65536, 365, 8) f32
    const float* w    = (const float*)d_in[1];  // (8, 1) f32
    // d_in[2] = weight_y: multiplied by an all-zeros matrix in the reference -> unused
    const float* bias = (const float*)d_in[3];  // (1,) f32
    float* out = (float*)d_out;                 // 65536 f32

    dim3 grid(kBatch / kRowsPerBlock);          // 512 blocks
    dim3 block(kWavesPerBlock * 32);            // 256 threads (8 wave32)
    ARX_LeafRiver_Qsim_kernel<<<grid, block, 0, stream>>>(x, w, bias, out);
}